// ContinuousConvolutionBase_81192061763734
// MI455X (gfx1250) — compile-verified
//
#include <hip/hip_runtime.h>
#include <cstdint>

// Problem constants from the reference.
#define B_  16
#define S_  4096
#define K_  9
#define C_  128          // INPUT_CHANNELS = 128 feature channels
#define SHIFT_MAX 8      // (KERNEL_SIZE-1)*DILATION
#define JROWS (S_ + SHIFT_MAX)   // 4104 "virtual" input rows per batch (tau in [-8, 4095])

typedef float f4 __attribute__((ext_vector_type(4)));

// One wave (32 lanes) owns one virtual input row (b, tau).
//   tau >= 0 : stage features[b, tau, :] (512 B) into LDS via the CDNA5 async
//              global->LDS DMA path, read it back to registers (ds_load_b128),
//              then scatter to the 9 shifted output rows with NT stores.
//   tau <  0 : row value is the zero-pad; store zeros to the (k, t<8-k) slots.
// Every output feature element is written exactly once.
__global__ __launch_bounds__(256) void scatter_features_kernel(
    const float* __restrict__ features, float* __restrict__ outF) {
  __shared__ float smem[8 * C_];            // 8 waves/block * one 512B row each
  const int wave = threadIdx.x >> 5;
  const int lane = threadIdx.x & 31;
  const int r = blockIdx.x * 8 + wave;
  if (r >= B_ * JROWS) return;
  const int b   = r / JROWS;
  const int tau = (r - b * JROWS) - SHIFT_MAX;

  f4 v = {0.0f, 0.0f, 0.0f, 0.0f};
  const int ldsIdx = wave * C_ + lane * 4;

  if (tau >= 0) {
    const float* src =
        features + ((size_t)(b * S_ + tau) * C_ + (size_t)lane * 4);
    const uint32_t ldsAddr = (uint32_t)(uintptr_t)(&smem[ldsIdx]);
    // CDNA5 async tensor-style copy: global -> LDS, 16B per lane (512B/wave),
    // tracked on ASYNCcnt; drain before consuming from LDS.
    asm volatile(
        "global_load_async_to_lds_b128 %0, %1, off\n\t"
        "s_wait_asynccnt 0"
        :
        : "v"(ldsAddr), "v"((uint64_t)(uintptr_t)src)
        : "memory");
    v = *reinterpret_cast<const f4*>(&smem[ldsIdx]);   // ds_load_b128
  }

#pragma unroll
  for (int k = 0; k < K_; ++k) {
    const int t = tau + (SHIFT_MAX - k);
    if ((unsigned)t < (unsigned)S_) {
      float* dst =
          outF + (((size_t)(b * K_ + k) * S_ + t) * C_ + (size_t)lane * 4);
      // Streaming 302MB output > 192MB L2: bias non-temporal.
      __builtin_nontemporal_store(v, reinterpret_cast<f4*>(dst));
    }
  }
}

// delta_times[b,k,t] = times[b,t] - (t >= 8-k ? times[b, t-(8-k)] : 0)
__global__ __launch_bounds__(256) void delta_times_kernel(
    const float* __restrict__ times, float* __restrict__ outDT) {
  const int idx = blockIdx.x * 256 + threadIdx.x;
  if (idx >= B_ * K_ * S_) return;
  const int t  = idx & (S_ - 1);
  const int bk = idx >> 12;      // / S_
  const int k  = bk % K_;
  const int b  = bk / K_;
  const int s  = SHIFT_MAX - k;
  const float tv = times[b * S_ + t];
  const float pv = (t >= s) ? times[b * S_ + (t - s)] : 0.0f;
  __builtin_nontemporal_store(tv - pv, &outDT[idx]);
}

extern "C" void kernel_launch(void* const* d_in, const int* in_sizes, int n_in,
                              void* d_out, int out_size, void* d_ws,
                              size_t ws_size, hipStream_t stream) {
  (void)in_sizes; (void)n_in; (void)out_size; (void)d_ws; (void)ws_size;
  const float* times    = (const float*)d_in[0];
  const float* features = (const float*)d_in[1];

  // d_out layout: [delta_times (B*K*S)] then [pre_conv_features (B*K*S*C)]
  float* outDT = (float*)d_out;
  float* outF  = outDT + (size_t)B_ * K_ * S_;

  const int rows = B_ * JROWS;                 // 65,664 waves
  dim3 grdA((rows + 7) / 8), blkA(256);        // 8 waves per block
  hipLaunchKernelGGL(scatter_features_kernel, grdA, blkA, 0, stream,
                     features, outF);

  const int nDT = B_ * K_ * S_;                // 589,824 elements
  dim3 grdB((nDT + 255) / 256), blkB(256);
  hipLaunchKernelGGL(delta_times_kernel, grdB, blkB, 0, stream, times, outDT);
}